// GCN_15496242004108
// MI455X (gfx1250) — compile-verified
//
#include <hip/hip_runtime.h>
#include <hip/hip_bf16.h>

#define IN_DIM 128
#define HID    128
#define OUTD   64

typedef __attribute__((ext_vector_type(2))) float v2f;
typedef __attribute__((ext_vector_type(8))) float v8f;

// ---------------- utility kernels ----------------

__global__ void fill_f32_kernel(float* __restrict__ p, int n, float v) {
    int i = blockIdx.x * blockDim.x + threadIdx.x;
    if (i < n) p[i] = v;
}

__global__ void deg_accum_kernel(const int* __restrict__ dst, float* __restrict__ deg, int nE) {
    int i = blockIdx.x * blockDim.x + threadIdx.x;
    if (i < nE) atomicAdd(deg + dst[i], 1.0f);
}

__global__ void recip_kernel(float* __restrict__ p, int n) {
    int i = blockIdx.x * blockDim.x + threadIdx.x;
    if (i < n) p[i] = 1.0f / p[i];   // deg >= 1 always (self-loop)
}

// vectorized copy: agg = x  (self-loop contribution); n4 = total floats / 4
__global__ void copy4_kernel(float4* __restrict__ dstp, const float4* __restrict__ srcp, int n4) {
    int i = blockIdx.x * blockDim.x + threadIdx.x;
    if (i < n4) dstp[i] = srcp[i];
}

// one wave (32 lanes) per edge; lane handles 4 consecutive feature dims
__global__ void scatter_add_kernel(const float* __restrict__ x,
                                   const int* __restrict__ src,
                                   const int* __restrict__ dst,
                                   float* __restrict__ agg, int nE) {
    int tid  = blockIdx.x * blockDim.x + threadIdx.x;
    int e    = tid >> 5;
    if (e >= nE) return;
    int lane = tid & 31;
    int s = src[e];
    int d = dst[e];
    const float4 v = *reinterpret_cast<const float4*>(x + (size_t)s * 128 + lane * 4);
    float* o = agg + (size_t)d * 128 + lane * 4;
    atomicAdd(o + 0, v.x);
    atomicAdd(o + 1, v.y);
    atomicAdd(o + 2, v.z);
    atomicAdd(o + 3, v.w);
}

// ---------------- fused SAGE GEMM ----------------
// out[r, c] = act( sum_k Aself[r,k] * Wself[k,c]
//                + sum_k (Aneigh[r,k]*invdeg[r]) * Wneigh[k,c] + bias[c] )
// One wave computes a 16x16 tile via V_WMMA_F32_16X16X4_F32 (K-step 4, K=2*128).
// NOUT is compile-time so all B addresses fold into immediate offsets.
// Requires: gridDim.x = N/16, blockDim.x = (NOUT/16)*32, N % 16 == 0.
template <int NOUT, bool RELU>
__global__ void sage_gemm_kernel(const float* __restrict__ Aself,   // [N,128]
                                 const float* __restrict__ Aneigh,  // [N,128]
                                 const float* __restrict__ invdeg,  // [N]
                                 const float* __restrict__ Wself,   // [128,NOUT]
                                 const float* __restrict__ Wneigh,  // [128,NOUT]
                                 const float* __restrict__ bias,    // [NOUT]
                                 float* __restrict__ out) {         // [N,NOUT]
    const int lane     = threadIdx.x & 31;
    const int wave     = threadIdx.x >> 5;
    const int row_base = blockIdx.x * 16;
    const int m        = lane & 15;
    const int hi       = lane >> 4;          // 0 for lanes 0-15, 1 for lanes 16-31
    const int koff     = hi << 1;            // A/B K sub-offset: 0 or 2
    const int row      = row_base + m;
    const int col      = wave * 16 + (lane & 15);

    const float  idg = invdeg[row];
    // A bases: element (m, k+koff) at offset k (immediate), pair load -> b64
    const float* aS  = Aself  + (size_t)row * 128 + koff;
    const float* aN  = Aneigh + (size_t)row * 128 + koff;
    // B bases: rows k+koff and k+koff+1, column col; k*NOUT folds to immediates
    const float* bS0 = Wself  + (size_t)koff * NOUT + col;
    const float* bS1 = bS0 + NOUT;
    const float* bN0 = Wneigh + (size_t)koff * NOUT + col;
    const float* bN1 = bN0 + NOUT;

    v8f c = {};

    // K = 0..127 : self path
    #pragma unroll
    for (int k = 0; k < 128; k += 4) {
        v2f a, b;
        a.x = aS[k];
        a.y = aS[k + 1];
        b.x = bS0[k * NOUT];
        b.y = bS1[k * NOUT];
        c = __builtin_amdgcn_wmma_f32_16x16x4_f32(false, a, false, b,
                                                  (short)0, c, false, false);
    }
    // K = 128..255 : neighbor-mean path (scaled by invdeg at load)
    #pragma unroll
    for (int k = 0; k < 128; k += 4) {
        v2f a, b;
        a.x = aN[k]     * idg;
        a.y = aN[k + 1] * idg;
        b.x = bN0[k * NOUT];
        b.y = bN1[k * NOUT];
        c = __builtin_amdgcn_wmma_f32_16x16x4_f32(false, a, false, b,
                                                  (short)0, c, false, false);
    }

    const float bv = bias[col];
    #pragma unroll
    for (int i = 0; i < 8; ++i) {
        // C/D layout: VGPR i, lanes 0-15 -> M=i ; lanes 16-31 -> M=8+i
        int   r = row_base + i + (hi << 3);
        float v = c[i] + bv;
        if (RELU) v = fmaxf(v, 0.0f);
        out[(size_t)r * NOUT + col] = v;
    }
}

// ---------------- launcher ----------------

extern "C" void kernel_launch(void* const* d_in, const int* in_sizes, int n_in,
                              void* d_out, int out_size, void* d_ws, size_t ws_size,
                              hipStream_t stream) {
    const float* feat = (const float*)d_in[0];
    const float* W1s  = (const float*)d_in[1];
    const float* W1n  = (const float*)d_in[2];
    const float* b1   = (const float*)d_in[3];
    const float* W2s  = (const float*)d_in[4];
    const float* W2n  = (const float*)d_in[5];
    const float* b2   = (const float*)d_in[6];
    const int*   esrc = (const int*)d_in[7];
    const int*   edst = (const int*)d_in[8];
    float*       outp = (float*)d_out;

    const int N = in_sizes[0] / IN_DIM;   // 100000
    const int E = in_sizes[7];            // 1600000

    // workspace layout (floats): deg[N] | agg[N*128] | h[N*128]
    float* deg = (float*)d_ws;
    float* agg = deg + N;
    float* h   = agg + (size_t)N * 128;

    const int TB = 256;
    const int nFeat  = N * 128;
    const int nFeat4 = nFeat / 4;
    const int scatterThreads = E * 32;

    // --- degree: 1 (self loop) + edge counts, then invert ---
    fill_f32_kernel<<<(N + TB - 1) / TB, TB, 0, stream>>>(deg, N, 1.0f);
    deg_accum_kernel<<<(E + TB - 1) / TB, TB, 0, stream>>>(edst, deg, E);
    recip_kernel<<<(N + TB - 1) / TB, TB, 0, stream>>>(deg, N);

    // --- layer 1 aggregation: agg = feat + scatter(feat[src] -> dst) ---
    copy4_kernel<<<(nFeat4 + TB - 1) / TB, TB, 0, stream>>>(
        (float4*)agg, (const float4*)feat, nFeat4);
    scatter_add_kernel<<<(scatterThreads + TB - 1) / TB, TB, 0, stream>>>(
        feat, esrc, edst, agg, E);

    // --- layer 1 GEMM: h = relu(feat@W1s + (agg/deg)@W1n + b1) ---
    sage_gemm_kernel<HID, true><<<N / 16, (HID / 16) * 32, 0, stream>>>(
        feat, agg, deg, W1s, W1n, b1, h);

    // --- layer 2 aggregation: agg = h + scatter(h[src] -> dst) ---
    copy4_kernel<<<(nFeat4 + TB - 1) / TB, TB, 0, stream>>>(
        (float4*)agg, (const float4*)h, nFeat4);
    scatter_add_kernel<<<(scatterThreads + TB - 1) / TB, TB, 0, stream>>>(
        h, esrc, edst, agg, E);

    // --- layer 2 GEMM: out = h@W2s + (agg/deg)@W2n + b2 ---
    sage_gemm_kernel<OUTD, false><<<N / 16, (OUTD / 16) * 32, 0, stream>>>(
        h, agg, deg, W2s, W2n, b2, outp);
}